// SemanticNumericAggregator_8718783611565
// MI455X (gfx1250) — compile-verified
//
#include <hip/hip_runtime.h>
#include <hip/hip_bf16.h>

typedef __attribute__((ext_vector_type(2))) float v2f;
typedef __attribute__((ext_vector_type(8))) float v8f;

// ---------------------------------------------------------------------------
// Generic f32 WMMA GEMM:  C[M,N] = A[M,Kd] * W[Kd,N] (+bias[N]) (opt. ReLU)
// One wave (32 lanes) per 16x16 output tile, K stepped by 4 via
// v_wmma_f32_16x16x4_f32.  All dims are multiples of 16 / 4 in this problem.
//
// VGPR layouts per CDNA5 ISA 7.12.2 (wave32):
//   A 16x4 :  lanes 0-15 -> M=lane,    a = {A[m][k+0], A[m][k+1]}
//             lanes16-31 -> M=lane-16, a = {A[m][k+2], A[m][k+3]}
//   B 4x16 :  lanes 0-15 -> N=lane,    b = {W[k+0][n], W[k+1][n]}
//             lanes16-31 -> N=lane-16, b = {W[k+2][n], W[k+3][n]}
//   C 16x16:  VGPR i: lanes 0-15 -> (M=i,   N=lane)
//                     lanes16-31 -> (M=i+8, N=lane-16)
// ---------------------------------------------------------------------------
__global__ __launch_bounds__(32) void wmma_gemm_f32(
    const float* __restrict__ A, const float* __restrict__ W,
    const float* __restrict__ bias, float* __restrict__ C,
    int Kd, int Nd, int relu)
{
    const int tilesN = Nd >> 4;
    const int tm = blockIdx.x / tilesN;
    const int tn = blockIdx.x % tilesN;
    const int lane = threadIdx.x;          // 0..31
    const int half = lane >> 4;            // 0 | 1
    const int l16  = lane & 15;
    const int m0 = tm << 4, n0 = tn << 4;

    // lane-private base pointers
    const float* Arow = A + (size_t)(m0 + l16) * Kd + 2 * half;  // -> A[m][k + 2*half]
    const float* Wcol = W + n0 + l16;                            // -> W[k][n]

    v8f c = {};
    for (int k = 0; k < Kd; k += 4) {
        v2f a, b;
        float2 av = *(const float2*)(Arow + k);   // A[m][k+2h], A[m][k+2h+1]
        a.x = av.x;
        a.y = av.y;
        b.x = Wcol[(size_t)(k + 2 * half)     * Nd];
        b.y = Wcol[(size_t)(k + 2 * half + 1) * Nd];
        c = __builtin_amdgcn_wmma_f32_16x16x4_f32(
                /*neg_a=*/false, a, /*neg_b=*/false, b,
                /*c_mod=*/(short)0, c, /*reuse_a=*/false, /*reuse_b=*/false);
    }

    const float bv = bias ? bias[n0 + l16] : 0.0f;
    float* Ccol = C + n0 + l16;
#pragma unroll
    for (int i = 0; i < 8; ++i) {
        const int m = m0 + i + 8 * half;
        float v = c[i] + bv;
        if (relu) v = fmaxf(v, 0.0f);
        Ccol[(size_t)m * Nd] = v;
    }
}

// ---------------------------------------------------------------------------
// Fused score -> softmax -> aggregation -> concat kernel.
// One block of 128 threads per (b,k) pair (1152 blocks).
//   logits[m] = sum_h relu(pre_s[b,m,h] + pre_x[b,k,h] + pre_q'[b,h]) * w2[h] + b2
//     (pre_q' already contains score_b1, folded in as the GEMM bias)
//   att = softmax_m(logits)
//   out[b,k, 0:128]  = x0[b,k,:]
//   out[b,k,128:256] = sum_m att[m] * h_s[b,m,:]
// ---------------------------------------------------------------------------
__global__ __launch_bounds__(128) void score_softmax_agg(
    const float* __restrict__ pre_s,   // [B,M,H]
    const float* __restrict__ pre_x,   // [B,K,H]
    const float* __restrict__ pre_q,   // [B,H]   (includes score_b1)
    const float* __restrict__ w2,      // [H]
    const float* __restrict__ b2,      // [1]
    const float* __restrict__ h_s,     // [B,M,DX]
    const float* __restrict__ x0,      // [B,K,DX]
    float* __restrict__ out)           // [B,K,2*DX]
{
    constexpr int M = 64, K = 36, H = 256, DX = 128;
    const int bk = blockIdx.x;
    const int b = bk / K, k = bk % K;
    const int t = threadIdx.x;

    __shared__ float pxq[H];
    __shared__ float w2s[H];
    __shared__ float att[M];
    __shared__ float smax, ssum;

    // Stage the per-(b,k), m-invariant H-vector and score_w2 into LDS.
    for (int h = t; h < H; h += 128) {
        pxq[h] = pre_x[((size_t)b * K + k) * H + h] + pre_q[(size_t)b * H + h];
        w2s[h] = w2[h];
    }
    __syncthreads();

    // Logits: one m per thread for t < 64.
    if (t < M) {
        const float* ps = pre_s + ((size_t)b * M + t) * H;
        float acc = 0.0f;
#pragma unroll 4
        for (int h = 0; h < H; ++h) {
            float v = ps[h] + pxq[h];
            acc = fmaf(fmaxf(v, 0.0f), w2s[h], acc);
        }
        att[t] = acc + b2[0];
    }
    __syncthreads();

    // Softmax over M=64 (wave32 shuffle reduction in wave 0).
    if (t < 32) {
        float v = fmaxf(att[t], att[t + 32]);
#pragma unroll
        for (int off = 16; off > 0; off >>= 1)
            v = fmaxf(v, __shfl_xor(v, off, 32));
        if (t == 0) smax = v;
    }
    __syncthreads();
    if (t < M) att[t] = __expf(att[t] - smax);
    __syncthreads();
    if (t < 32) {
        float v = att[t] + att[t + 32];
#pragma unroll
        for (int off = 16; off > 0; off >>= 1)
            v += __shfl_xor(v, off, 32);
        if (t == 0) ssum = v;
    }
    __syncthreads();
    const float inv = 1.0f / ssum;

    // Aggregation + concat: DX == blockDim.x, one output feature per thread.
    float* orow = out + ((size_t)b * K + k) * (2 * DX);
    orow[t] = x0[((size_t)b * K + k) * DX + t];

    float acc = 0.0f;
#pragma unroll 4
    for (int m = 0; m < M; ++m)
        acc = fmaf(att[m], h_s[((size_t)b * M + m) * DX + t], acc);
    orow[DX + t] = acc * inv;
}

extern "C" void kernel_launch(void* const* d_in, const int* in_sizes, int n_in,
                              void* d_out, int out_size, void* d_ws, size_t ws_size,
                              hipStream_t stream) {
    (void)in_sizes; (void)n_in; (void)out_size; (void)ws_size;
    const float* s2  = (const float*)d_in[0];   // [32,64,512]
    const float* x0  = (const float*)d_in[1];   // [32,36,128]
    const float* q   = (const float*)d_in[2];   // [32,512]
    const float* w1  = (const float*)d_in[3];   // [1152,256]
    const float* b1  = (const float*)d_in[4];   // [256]
    const float* w2  = (const float*)d_in[5];   // [256,1]
    const float* b2  = (const float*)d_in[6];   // [1]
    const float* hw1 = (const float*)d_in[7];   // [512,256]
    const float* hb1 = (const float*)d_in[8];   // [256]
    const float* hw2 = (const float*)d_in[9];   // [256,128]
    const float* hb2 = (const float*)d_in[10];  // [128]
    float* out = (float*)d_out;

    // Workspace layout (f32 elements)
    float* ws    = (float*)d_ws;
    float* pre_s = ws;                 // 32*64*256  = 524288
    float* pre_x = ws + 524288;        // 32*36*256  = 294912
    float* pre_q = ws + 819200;        // 32*256     = 8192
    float* h1    = ws + 827392;        // 32*64*256  = 524288
    float* h_s   = ws + 1351680;       // 32*64*128  = 262144

    dim3 blk(32);
    // pre_s = s2 @ w1[0:512]        : M=2048, Kd=512, N=256 -> 128*16 tiles
    wmma_gemm_f32<<<2048, blk, 0, stream>>>(s2, w1,              nullptr, pre_s, 512, 256, 0);
    // pre_x = x0 @ w1[512:640]      : M=1152, Kd=128, N=256 -> 72*16 tiles
    wmma_gemm_f32<<<1152, blk, 0, stream>>>(x0, w1 + 512 * 256,  nullptr, pre_x, 128, 256, 0);
    // pre_q = q @ w1[640:1152] + b1 : M=32,   Kd=512, N=256 -> 2*16 tiles (score_b1 folded here)
    wmma_gemm_f32<<<32,   blk, 0, stream>>>(q,  w1 + 640 * 256,  b1,      pre_q, 512, 256, 0);
    // h1 = relu(s2 @ hw1 + hb1)     : M=2048, Kd=512, N=256
    wmma_gemm_f32<<<2048, blk, 0, stream>>>(s2, hw1,             hb1,     h1,    512, 256, 1);
    // h_s = h1 @ hw2 + hb2          : M=2048, Kd=256, N=128 -> 128*8 tiles
    wmma_gemm_f32<<<1024, blk, 0, stream>>>(h1, hw2,             hb2,     h_s,   256, 128, 0);

    // Fused score / softmax / aggregation / concat: one block per (b,k)
    score_softmax_agg<<<32 * 36, 128, 0, stream>>>(pre_s, pre_x, pre_q, w2, b2, h_s, x0, out);
}